// SparseBEVAttention_60756607369848
// MI455X (gfx1250) — compile-verified
//
#include <hip/hip_runtime.h>
#include <math.h>

typedef __attribute__((ext_vector_type(2))) float v2f;
typedef __attribute__((ext_vector_type(8))) float v8f;

constexpr int kB   = 8;
constexpr int kNQ  = 4096;
constexpr int kNK  = 4096;
constexpr int kD   = 2;
constexpr int kH   = 64;
constexpr int kCV  = 64;
constexpr int kTOP = 16;

constexpr int kWAVES  = 8;     // waves per block
constexpr int kQPW    = 32;    // queries per wave (two 16-row WMMA tiles)
constexpr int kQPB    = kWAVES * kQPW;  // 256 queries per block
constexpr int kKCHUNK = 512;   // keys staged in LDS per chunk
constexpr int kTILES  = kKCHUNK / 16;

__global__ __launch_bounds__(256)
void sparse_bev_knn_attn(const float* __restrict__ q,
                         const float* __restrict__ k,
                         const float* __restrict__ v,
                         const float* __restrict__ wq,   // (H,2)
                         const float* __restrict__ bq,   // (H)
                         const float* __restrict__ wk,   // (H,2)
                         const float* __restrict__ bk,   // (H)
                         float* __restrict__ out)        // (B,NQ,CV)
{
    // LDS staging
    __shared__ float4 kbuf[kKCHUNK];                 // {-2kx,-2ky,k2,1}
    __shared__ float  tbuf[kWAVES][kQPW][17];        // WMMA tile transpose (padded)
    __shared__ float  abuf[kWAVES][kQPW][kTOP];      // softmax weights
    __shared__ int    ibuf[kWAVES][kQPW][kTOP];      // knn indices
    __shared__ float  pwq0[kH], pwq1[kH], pbq[kH], pwk0[kH], pwk1[kH], pbk[kH];

    const int tid  = threadIdx.x;
    const int wave = tid >> 5;
    const int lane = tid & 31;
    const int m    = lane & 15;
    const bool hi  = lane >= 16;

    const int b    = blockIdx.x / (kNQ / kQPB);
    const int qblk = (blockIdx.x % (kNQ / kQPB)) * kQPB;
    const int q0   = qblk + wave * kQPW;   // first query of this wave
    const int myq  = q0 + lane;            // query owned by this lane

    // projection params -> LDS (synced by first __syncthreads in chunk loop)
    if (tid < kH) {
        pwq0[tid] = wq[tid * 2 + 0];
        pwq1[tid] = wq[tid * 2 + 1];
        pbq[tid]  = bq[tid];
        pwk0[tid] = wk[tid * 2 + 0];
        pwk1[tid] = wk[tid * 2 + 1];
        pbk[tid]  = bk[tid];
    }

    // owned query coordinates (for the attention phase)
    const float qx_o = q[((size_t)b * kNQ + myq) * kD + 0];
    const float qy_o = q[((size_t)b * kNQ + myq) * kD + 1];

    // ---- Build the two A operands (16x4 f32 each): row = [qx, qy, 1, q^2] ----
    // Layout (ISA 7.12.2, 32-bit A 16x4): VGPR0 lanes0-15 K=0, lanes16-31 K=2;
    //                                     VGPR1 lanes0-15 K=1, lanes16-31 K=3.
    const float qx0 = q[((size_t)b * kNQ + q0 + m) * kD + 0];
    const float qy0 = q[((size_t)b * kNQ + q0 + m) * kD + 1];
    const float qx1 = q[((size_t)b * kNQ + q0 + 16 + m) * kD + 0];
    const float qy1 = q[((size_t)b * kNQ + q0 + 16 + m) * kD + 1];
    v2f a0, a1;
    a0.x = hi ? 1.0f : qx0;
    a0.y = hi ? (qx0 * qx0 + qy0 * qy0) : qy0;
    a1.x = hi ? 1.0f : qx1;
    a1.y = hi ? (qx1 * qx1 + qy1 * qy1) : qy1;

    // register-resident sorted top-16 (ascending squared distance)
    float dist[kTOP];
    int   idx[kTOP];
#pragma unroll
    for (int j = 0; j < kTOP; ++j) { dist[j] = 3.0e38f; idx[j] = 0; }

    // =================== Phase A: WMMA distances + top-k ===================
    for (int ch = 0; ch < kNK; ch += kKCHUNK) {
        __syncthreads();   // kbuf reuse guard (+ covers proj-param writes)
        for (int t = tid; t < kKCHUNK; t += 256) {
            const float kx = k[((size_t)b * kNK + ch + t) * kD + 0];
            const float ky = k[((size_t)b * kNK + ch + t) * kD + 1];
            kbuf[t] = make_float4(-2.0f * kx, -2.0f * ky, kx * kx + ky * ky, 1.0f);
        }
        __syncthreads();

        for (int t = 0; t < kTILES; ++t) {
            // B operand (4x16 f32): col n = [-2kx, -2ky, k2, 1]^T
            // VGPR0 lanes0-15 K=0, lanes16-31 K=2; VGPR1 lanes0-15 K=1, lanes16-31 K=3.
            const float4 kbv = kbuf[t * 16 + m];
            v2f bm;
            bm.x = hi ? kbv.z : kbv.x;
            bm.y = hi ? kbv.w : kbv.y;

            v8f cz = {};
            // d2[16x16] = A(16x4) * B(4x16): one WMMA per 16-query tile
            v8f d0 = __builtin_amdgcn_wmma_f32_16x16x4_f32(false, a0, false, bm,
                                                           (short)0, cz, false, false);
            v8f d1 = __builtin_amdgcn_wmma_f32_16x16x4_f32(false, a1, false, bm,
                                                           (short)0, cz, false, false);

            // transpose D tiles through LDS: row = query-in-wave, col = key-in-tile
            const int rbase = hi ? 8 : 0;
#pragma unroll
            for (int r = 0; r < 8; ++r) {
                tbuf[wave][rbase + r][m]      = d0[r];
                tbuf[wave][16 + rbase + r][m] = d1[r];
            }
            // same-wave cross-lane handoff: LDS is in-order per wave; fence + wait
            __builtin_amdgcn_wave_barrier();
            asm volatile("s_wait_dscnt 0" ::: "memory");
            __builtin_amdgcn_wave_barrier();

            // lane L owns query q0+L: insert 16 candidates into sorted top-16
            const int kb0 = ch + t * 16;
#pragma unroll
            for (int j = 0; j < 16; ++j) {
                const float dd = tbuf[wave][lane][j];
                if (dd < dist[kTOP - 1]) {
                    dist[kTOP - 1] = dd;
                    idx[kTOP - 1]  = kb0 + j;
#pragma unroll
                    for (int s = kTOP - 1; s >= 1; --s) {
                        if (dist[s] < dist[s - 1]) {
                            const float td = dist[s]; dist[s] = dist[s - 1]; dist[s - 1] = td;
                            const int   ti = idx[s];  idx[s]  = idx[s - 1];  idx[s - 1]  = ti;
                        }
                    }
                }
            }
            __builtin_amdgcn_wave_barrier();
            asm volatile("" ::: "memory");  // keep loads before next tile's stores
        }
    }

    // =================== Phase B: projected attention ===================
    // q_emb[h] = Wq[h]*(qx,qy)+bq[h]; score_j = a*kx_j + bcf*ky_j + c (H collapsed)
    float ca = 0.0f, cbf = 0.0f, cc = 0.0f;
#pragma unroll 8
    for (int h = 0; h < kH; ++h) {
        const float qe = fmaf(pwq0[h], qx_o, fmaf(pwq1[h], qy_o, pbq[h]));
        ca  = fmaf(qe, pwk0[h], ca);
        cbf = fmaf(qe, pwk1[h], cbf);
        cc  = fmaf(qe, pbk[h], cc);
    }

    float sc[kTOP];
    float smax = -3.0e38f;
#pragma unroll
    for (int j = 0; j < kTOP; ++j) {
        const int key  = idx[j];
        const float kx = k[((size_t)b * kNK + key) * kD + 0];
        const float ky = k[((size_t)b * kNK + key) * kD + 1];
        sc[j] = fmaf(ca, kx, fmaf(cbf, ky, cc));
        smax  = fmaxf(smax, sc[j]);
    }
    float ssum = 0.0f;
#pragma unroll
    for (int j = 0; j < kTOP; ++j) { sc[j] = __expf(sc[j] - smax); ssum += sc[j]; }
    const float rs = 1.0f / ssum;
#pragma unroll
    for (int j = 0; j < kTOP; ++j) {
        abuf[wave][lane][j] = sc[j] * rs;
        ibuf[wave][lane][j] = idx[j];
    }
    __builtin_amdgcn_wave_barrier();
    asm volatile("s_wait_dscnt 0" ::: "memory");
    __builtin_amdgcn_wave_barrier();

    // cooperative weighted gather of v: 32 lanes cover 64 channels (float2/lane)
    const float2* __restrict__ v2p = reinterpret_cast<const float2*>(v);
    float2* __restrict__ o2p = reinterpret_cast<float2*>(out);
    for (int qq = 0; qq < kQPW; ++qq) {
        float2 acc = make_float2(0.0f, 0.0f);
#pragma unroll
        for (int j = 0; j < kTOP; ++j) {
            const float w  = abuf[wave][qq][j];   // LDS broadcast
            const int  key = ibuf[wave][qq][j];
            const float2 vv = v2p[((size_t)b * kNK + key) * (kCV / 2) + lane];
            acc.x = fmaf(w, vv.x, acc.x);
            acc.y = fmaf(w, vv.y, acc.y);
        }
        o2p[((size_t)b * kNQ + (q0 + qq)) * (kCV / 2) + lane] = acc;
    }
}

extern "C" void kernel_launch(void* const* d_in, const int* in_sizes, int n_in,
                              void* d_out, int out_size, void* d_ws, size_t ws_size,
                              hipStream_t stream) {
    const float* q  = (const float*)d_in[0];
    const float* k  = (const float*)d_in[1];
    const float* v  = (const float*)d_in[2];
    const float* wq = (const float*)d_in[3];
    const float* bq = (const float*)d_in[4];
    const float* wk = (const float*)d_in[5];
    const float* bk = (const float*)d_in[6];
    // d_in[7] = top_k (compile-time 16 here)
    float* out = (float*)d_out;

    const int blocks = kB * (kNQ / kQPB);   // 8 * 16 = 128
    sparse_bev_knn_attn<<<blocks, 256, 0, stream>>>(q, k, v, wq, bq, wk, bk, out);

    (void)in_sizes; (void)n_in; (void)out_size; (void)d_ws; (void)ws_size;
}